// LLFunctional_79697413144888
// MI455X (gfx1250) — compile-verified
//
#include <hip/hip_runtime.h>
#include <hip/hip_bf16.h>
#include <stddef.h>
#include <stdint.h>

// ---------------------------------------------------------------------------
// Per-test-point weighted ridge regression (local linear regression with RBF
// weights) for MI455X / gfx1250, built around V_WMMA_F32_16X16X4_F32 and the
// CDNA5 async-to-LDS path.
//
// Pipeline:
//   1) build X~ (ones column prepended, dp=32), k-pair-interleaved X~^T,
//      row sumsq, and P2 (k-pair-interleaved):
//        P[n, 0:1024]    = outer(X~[n], X~[n]) flattened
//        P[n, 1024:1056] = X~[n] * y[n]          (cols 1056:1088 zero pad)
//   2) K = exp(-sqdist(X~te, X~tr) / (2 ls^2))  via WMMA f32 Gram + fused exp
//   3) C = K @ P  (4096x2048 @ 2048x1088)       via WMMA f32, 16x64 wave tiles
//   4) per-test 32x32 ridge solve: async-DMA the augmented system into LDS
//      (global_load_async_to_lds_b128 + s_wait_asynccnt), row-per-lane
//      Gauss-Jordan, shuffle-reduce the prediction dot product.
// ---------------------------------------------------------------------------

typedef __attribute__((ext_vector_type(2))) float v2f;
typedef __attribute__((ext_vector_type(8))) float v8f;

#define N_TRAIN 2048
#define N_TEST  4096
#define D_IN    31
#define DP      32
#define NC      1056          // 1024 (XWX) + 32 (XWy)
#define NCP     1088          // padded to 17 strips of 64 WMMA columns
#define RIDGE   1e-6f

static __device__ __forceinline__ v8f wmma_f32(v2f a, v2f b, v8f c) {
  // V_WMMA_F32_16X16X4_F32 : D(16x16,f32) = A(16x4,f32) x B(4x16,f32) + C
  return __builtin_amdgcn_wmma_f32_16x16x4_f32(
      /*neg_a=*/false, a, /*neg_b=*/false, b,
      /*c_mod=*/(short)0, c, /*reuse_a=*/false, /*reuse_b=*/false);
}

// --------------------------------------------------------------------------
// Kernel 1: X~ = [1 | X], row sumsq, optional k-pair-interleaved transpose
// XtP[(k>>1)*(2n) + 2r + (k&1)] = X~[r,k]   (B-fragment float2 loads later).
// --------------------------------------------------------------------------
__global__ void build_xtild_kernel(const float* __restrict__ X,   // [n, 31]
                                   float* __restrict__ Xt,        // [n, 32]
                                   float* __restrict__ XtP,       // [32, n] pair-ilv or nullptr
                                   float* __restrict__ norms,     // [n]
                                   int n) {
  int r = blockIdx.x * blockDim.x + threadIdx.x;
  if (r >= n) return;
  float s = 1.0f;                       // ones column contributes 1
  Xt[(size_t)r * DP + 0] = 1.0f;
  if (XtP) XtP[2 * r] = 1.0f;           // k=0 -> pair 0, slot 0
  #pragma unroll
  for (int j = 0; j < D_IN; ++j) {
    float v = X[(size_t)r * D_IN + j];
    int k = 1 + j;
    Xt[(size_t)r * DP + k] = v;
    if (XtP) XtP[(size_t)(k >> 1) * (2 * n) + 2 * r + (k & 1)] = v;
    s += v * v;
  }
  norms[r] = s;
}

// --------------------------------------------------------------------------
// Kernel 2: build pair-interleaved P2:
//   elem(n, c) at P2[(n>>1)*2*NCP + 2*c + (n&1)]
//   c<1024: X~[n,c>>5]*X~[n,c&31];  c<1056: X~[n,c-1024]*y[n];  else 0.
// --------------------------------------------------------------------------
__global__ void build_p_kernel(const float* __restrict__ Xtr,  // [2048, 32]
                               const float* __restrict__ Y,    // [2048]
                               float* __restrict__ P2) {       // [1024][NCP][2]
  int n = blockIdx.x;
  __shared__ float x[DP];
  if (threadIdx.x < DP) x[threadIdx.x] = Xtr[(size_t)n * DP + threadIdx.x];
  __syncthreads();
  float yn = Y[n];
  float* base = P2 + (size_t)(n >> 1) * (2 * NCP) + (n & 1);
  for (int c = threadIdx.x; c < NCP; c += blockDim.x) {
    float v;
    if (c < DP * DP)      v = x[c >> 5] * x[c & 31];
    else if (c < NC)      v = x[c - DP * DP] * yn;
    else                  v = 0.0f;
    base[2 * c] = v;
  }
}

// --------------------------------------------------------------------------
// Kernel 3: K[t, n] = exp(-(||xt||^2 + ||xn||^2 - 2 xt.xn) / (2 ls^2))
// One wave per 16x16 tile; Gram via 8x WMMA f32 16x16x4 over K=32,
// exp applied directly on the accumulator registers.
// --------------------------------------------------------------------------
__global__ void gram_rbf_kernel(const float* __restrict__ Xte,   // [4096, 32]
                                const float* __restrict__ XtrP,  // [16][2048][2]
                                const float* __restrict__ nte,   // [4096]
                                const float* __restrict__ ntr,   // [2048]
                                const float* __restrict__ logls, // [1]
                                float* __restrict__ Kout) {      // [4096, 2048]
  const int mBase = blockIdx.x * 16;
  const int nBase = blockIdx.y * 16;
  const int lane  = threadIdx.x & 31;
  const int l     = lane & 15;
  const int hi    = lane >> 4;

  v8f acc = {};
  const float* ap = Xte + (size_t)(mBase + l) * DP + 2 * hi;
  const float* bp = XtrP + (size_t)hi * (2 * N_TRAIN) + 2 * (nBase + l);
  #pragma unroll
  for (int k0 = 0; k0 < DP; k0 += 4) {
    v2f a = *(const v2f*)(ap + k0);                               // A(row, k0+2hi +{0,1})
    v2f b = *(const v2f*)(bp + (size_t)(k0 >> 1) * (2 * N_TRAIN)); // B(k0+2hi +{0,1}, col)
    acc = wmma_f32(a, b, acc);
  }

  float ls      = __expf(*logls);
  float inv2ls2 = 1.0f / (2.0f * ls * ls);
  float nn      = ntr[nBase + l];
  #pragma unroll
  for (int j = 0; j < 8; ++j) {
    int   row = mBase + j + 8 * hi;
    float sq  = nte[row] + nn - 2.0f * acc[j];
    Kout[(size_t)row * N_TRAIN + nBase + l] = __expf(-sq * inv2ls2);
  }
}

// --------------------------------------------------------------------------
// Kernel 4: C[4096 x 1088] = K[4096 x 2048] @ P[2048 x 1088]  (dominant GEMM)
// One wave per 16(M) x 64(N) strip: four accumulators share each A fragment;
// every fragment load is one aligned float2 (b64).  4 WMMA : 5 loads per step.
// --------------------------------------------------------------------------
__global__ void xwx_gemm_kernel(const float* __restrict__ Kmat, // [4096, 2048]
                                const float* __restrict__ P2,   // [1024][NCP][2]
                                float* __restrict__ C) {        // [4096, NCP]
  const int mBase = blockIdx.x * 16;
  const int nBase = blockIdx.y * 64;
  const int lane  = threadIdx.x & 31;
  const int l     = lane & 15;
  const int hi    = lane >> 4;

  v8f acc0 = {}, acc1 = {}, acc2 = {}, acc3 = {};
  const float* arow = Kmat + (size_t)(mBase + l) * N_TRAIN + 2 * hi;
  const float* bcol = P2 + (size_t)hi * (2 * NCP) + 2 * (nBase + l);

  #pragma unroll 4
  for (int k0 = 0; k0 < N_TRAIN; k0 += 4) {
    v2f a = *(const v2f*)(arow + k0);
    const float* bp = bcol + (size_t)(k0 >> 1) * (2 * NCP);
    v2f b0 = *(const v2f*)(bp + 0);
    v2f b1 = *(const v2f*)(bp + 32);
    v2f b2 = *(const v2f*)(bp + 64);
    v2f b3 = *(const v2f*)(bp + 96);
    acc0 = wmma_f32(a, b0, acc0);
    acc1 = wmma_f32(a, b1, acc1);
    acc2 = wmma_f32(a, b2, acc2);
    acc3 = wmma_f32(a, b3, acc3);
  }

  #pragma unroll
  for (int j = 0; j < 8; ++j) {
    int row = mBase + j + 8 * hi;
    float* crow = C + (size_t)row * NCP + nBase + l;
    crow[0]  = acc0[j];
    crow[16] = acc1[j];
    crow[32] = acc2[j];
    crow[48] = acc3[j];
  }
}

// --------------------------------------------------------------------------
// Kernel 5: per-test 32x32 ridge solve + prediction.
// One wave per test point, 8 tests per 256-thread block.
// Augmented [A | b] is DMA'd into LDS via CDNA5 async-to-LDS loads
// (ASYNCcnt-tracked), then row-per-lane Gauss-Jordan (SPD + ridge, no pivot).
// LDS row stride = 36 floats = 144 B so every 16 B chunk stays 16 B aligned.
// --------------------------------------------------------------------------
__global__ void solve_predict_kernel(const float* __restrict__ C,   // [4096, NCP]
                                     const float* __restrict__ Xte, // [4096, 32]
                                     float* __restrict__ out) {     // [4096]
  const int wave = threadIdx.x >> 5;   // 0..7
  const int lane = threadIdx.x & 31;   // row index owned by this lane
  const int t    = blockIdx.x * 8 + wave;

  __shared__ __align__(16) float sm[8][DP][DP + 4];  // col 32 = rhs, pad->36

  const float* crow = C + (size_t)t * NCP;
  float xr = Xte[(size_t)t * DP + lane];

  // Async DMA this lane's matrix row (32 floats) + rhs element into LDS.
  {
    uint32_t ldsRow = (uint32_t)(uintptr_t)(&sm[wave][lane][0]);
    uint64_t gRow   = (uint64_t)(uintptr_t)(crow + (size_t)lane * DP);
    #pragma unroll
    for (int q = 0; q < 8; ++q) {
      asm volatile("global_load_async_to_lds_b128 %0, %1, off"
                   :: "v"(ldsRow + q * 16), "v"(gRow + (uint64_t)q * 16)
                   : "memory");
    }
    uint32_t ldsRhs = (uint32_t)(uintptr_t)(&sm[wave][lane][DP]);
    uint64_t gRhs   = (uint64_t)(uintptr_t)(crow + DP * DP + lane);
    asm volatile("global_load_async_to_lds_b32 %0, %1, off"
                 :: "v"(ldsRhs), "v"(gRhs) : "memory");
    asm volatile("s_wait_asynccnt 0x0" ::: "memory");
  }
  sm[wave][lane][lane] += RIDGE;
  __syncthreads();

  for (int k = 0; k < DP; ++k) {
    if (lane != k) {
      float f = sm[wave][lane][k] / sm[wave][k][k];
      for (int j = k; j <= DP; ++j)
        sm[wave][lane][j] -= f * sm[wave][k][j];
    }
    __syncthreads();
  }

  float beta = sm[wave][lane][DP] / sm[wave][lane][lane];
  float val  = xr * beta;
  #pragma unroll
  for (int off = 16; off > 0; off >>= 1)
    val += __shfl_xor(val, off, 32);
  if (lane == 0) out[t] = val;
}

// --------------------------------------------------------------------------
// Host-side launch.  Workspace layout (floats), total ~61.4 MB:
//   Xtr[2048*32] XtrP[32*2048] Xte[4096*32] ntr[2048] nte[4096]
//   P2[2048*1088] K[4096*2048] C[4096*1088]
// --------------------------------------------------------------------------
extern "C" void kernel_launch(void* const* d_in, const int* in_sizes, int n_in,
                              void* d_out, int out_size, void* d_ws, size_t ws_size,
                              hipStream_t stream) {
  const float* Ytrain = (const float*)d_in[0];   // [2048]
  const float* Xtrain = (const float*)d_in[1];   // [2048, 31]
  const float* Xtest  = (const float*)d_in[2];   // [4096, 31]
  const float* logls  = (const float*)d_in[3];   // [1]
  float* out = (float*)d_out;                    // [4096]

  float* ws   = (float*)d_ws;
  float* Xtr  = ws;                                        // 65536
  float* XtrP = Xtr  + (size_t)N_TRAIN * DP;               // 65536
  float* Xte  = XtrP + (size_t)DP * N_TRAIN;               // 131072
  float* ntr  = Xte  + (size_t)N_TEST * DP;                // 2048
  float* nte  = ntr  + N_TRAIN;                            // 4096
  float* P2   = nte  + N_TEST;                             // 2228224
  float* Km   = P2   + (size_t)N_TRAIN * NCP;              // 8388608
  float* Cm   = Km   + (size_t)N_TEST * N_TRAIN;           // 4456448

  // 1) X~, pair-interleaved transpose, norms
  build_xtild_kernel<<<(N_TRAIN + 127) / 128, 128, 0, stream>>>(
      Xtrain, Xtr, XtrP, ntr, N_TRAIN);
  build_xtild_kernel<<<(N_TEST + 127) / 128, 128, 0, stream>>>(
      Xtest, Xte, nullptr, nte, N_TEST);

  // 2) P2 = [outer(X~,X~) | X~*y | 0-pad], k-pair interleaved
  build_p_kernel<<<N_TRAIN, 256, 0, stream>>>(Xtr, Ytrain, P2);

  // 3) K = RBF gram (WMMA f32 + fused exp).  Tiles: 256 x 128.
  gram_rbf_kernel<<<dim3(N_TEST / 16, N_TRAIN / 16), 32, 0, stream>>>(
      Xte, XtrP, nte, ntr, logls, Km);

  // 4) C = K @ P (WMMA f32).  Strips: 256 x 17 (each 16x64).
  xwx_gemm_kernel<<<dim3(N_TEST / 16, NCP / 64), 32, 0, stream>>>(Km, P2, Cm);

  // 5) per-test ridge solve + prediction (async-to-LDS staging).
  solve_predict_kernel<<<N_TEST / 8, 256, 0, stream>>>(Cm, Xte, out);
}